// attn_28681791603139
// MI455X (gfx1250) — compile-verified
//
#include <hip/hip_runtime.h>
#include <hip/hip_bf16.h>

// ---------------------------------------------------------------------------
// GAT layer for MI455X (gfx1250, wave32).
//   Wh  = lrelu(X @ Ww^T + Wb)          (dense GEMM -> V_WMMA_F32_16X16X4_F32)
//   res =        X @ rw^T + rb          (dense GEMM -> V_WMMA_F32_16X16X4_F32)
//   sd[n] = Wh[n].aw[:64], ss[n] = Wh[n].aw[64:]
//   per edge: w = exp(-lrelu(sd[dst]+ss[src]+ab)); scatter w, w*Wh[src] by dst
//   out = lrelu(agg/(rowsum+eps) + res)
// ---------------------------------------------------------------------------

#define B_     4
#define N_     10000
#define E_     320000
#define IN_DIM 128
#define OUT_DIM 64
#define BN     (B_ * N_)            // 40000 rows, divisible by 16
#define ALPHA  0.2f
#define EPS_   9e-15f

typedef float v2f __attribute__((ext_vector_type(2)));
typedef float v8f __attribute__((ext_vector_type(8)));

__device__ __forceinline__ float lrelu(float x) { return x > 0.0f ? x : ALPHA * x; }

// ---------------------------------------------------------------------------
// Kernel 1: tiled GEMM via f32 WMMA. One wave computes one 16x16 output tile.
// Block = 256 threads = 8 waves: waves 0..3 -> Wh col-tiles 0..3,
//                                waves 4..7 -> res col-tiles 0..3.
// Grid.x = BN/16 = 2500 row tiles. No divergence => EXEC all ones for WMMA.
// ---------------------------------------------------------------------------
__global__ __launch_bounds__(256) void gat_gemm_kernel(
    const float* __restrict__ X,    // (BN, 128)
    const float* __restrict__ Ww,   // (64, 128)
    const float* __restrict__ Wb,   // (64)
    const float* __restrict__ rw,   // (64, 128)
    const float* __restrict__ rb,   // (64)
    float* __restrict__ Wh,         // (BN, 64)
    float* __restrict__ res)        // (BN, 64)
{
    const int lane    = threadIdx.x & 31;
    const int wave    = threadIdx.x >> 5;     // 0..7
    const int colTile = wave & 3;             // 0..3  (16 cols each)
    const int isRes   = wave >> 2;            // 0 -> Ww path, 1 -> rw path
    const int rowTile = blockIdx.x;           // 0..2499

    const int r    = lane & 15;               // 0..15
    const int half = lane >> 4;               // 0 or 1

    const float* __restrict__ Wm = isRes ? rw : Ww;

    // A: X rows (16 x 4 f32 per step). Lane r,half holds A[r][half*2 + j].
    const float* xrow = X  + (size_t)(rowTile * 16 + r) * IN_DIM;
    // B: Wm^T (4 x 16 f32 per step). Lane r,half holds B[half*2+j][r] = Wm[r][half*2+j].
    const float* wrow = Wm + (size_t)(colTile * 16 + r) * IN_DIM;

    v8f acc = {0.f, 0.f, 0.f, 0.f, 0.f, 0.f, 0.f, 0.f};

#pragma unroll
    for (int k0 = 0; k0 < IN_DIM; k0 += 4) {
        v2f a, b;
        a.x = xrow[k0 + half * 2 + 0];
        a.y = xrow[k0 + half * 2 + 1];
        b.x = wrow[k0 + half * 2 + 0];
        b.y = wrow[k0 + half * 2 + 1];
        // D = A(16x4) * B(4x16) + C ; full fp32
        acc = __builtin_amdgcn_wmma_f32_16x16x4_f32(
            /*neg_a=*/false, a, /*neg_b=*/false, b,
            /*c_mod=*/(short)0, acc, /*reuse_a=*/false, /*reuse_b=*/false);
    }

    // C/D layout: VGPR v at lane -> row M = v + 8*half, col Nc = r.
    const int col = colTile * 16 + r;
#pragma unroll
    for (int v = 0; v < 8; ++v) {
        const int row = rowTile * 16 + v + 8 * half;
        float val = acc[v];
        if (isRes) {
            val += rb[col];
            res[(size_t)row * OUT_DIM + col] = val;
        } else {
            val += Wb[col];
            Wh[(size_t)row * OUT_DIM + col] = lrelu(val);
        }
    }
}

// ---------------------------------------------------------------------------
// Kernel 2: per-node attention dot products + rowsum zeroing.
// ---------------------------------------------------------------------------
__global__ __launch_bounds__(256) void gat_node_score_kernel(
    const float* __restrict__ Wh, const float* __restrict__ aw,
    float* __restrict__ sd, float* __restrict__ ss, float* __restrict__ rowsum)
{
    const int i = blockIdx.x * blockDim.x + threadIdx.x;
    if (i >= BN) return;
    const float* row = Wh + (size_t)i * OUT_DIM;
    float a = 0.f, s = 0.f;
#pragma unroll
    for (int c = 0; c < OUT_DIM; ++c) {
        const float v = row[c];
        a += v * aw[c];
        s += v * aw[OUT_DIM + c];
    }
    sd[i] = a;
    ss[i] = s;
    rowsum[i] = 0.f;
}

// ---------------------------------------------------------------------------
// Kernel 3: zero the agg accumulator.
// ---------------------------------------------------------------------------
__global__ __launch_bounds__(256) void gat_zero_kernel(float* __restrict__ p, int n)
{
    const int i = blockIdx.x * blockDim.x + threadIdx.x;
    if (i < n) p[i] = 0.f;
}

// ---------------------------------------------------------------------------
// Kernel 4: edge scatter. One wave per (edge, batch) pair.
// Lane 0 accumulates rowsum; 32 lanes accumulate 64 components of w*Wh[src].
// Grid sized exactly: E*B waves, no guard (full EXEC, coalesced 256B row reads).
// ---------------------------------------------------------------------------
__global__ __launch_bounds__(256) void gat_edge_kernel(
    const int* __restrict__ edge_dst, const int* __restrict__ edge_src,
    const float* __restrict__ sd, const float* __restrict__ ss,
    const float* __restrict__ ab, const float* __restrict__ Wh,
    float* __restrict__ rowsum, float* __restrict__ agg)
{
    const int lane = threadIdx.x & 31;
    const long long widx =
        (long long)blockIdx.x * (blockDim.x >> 5) + (threadIdx.x >> 5);
    const int e = (int)(widx >> 2);   // edge index   (B_ == 4)
    const int b = (int)(widx & 3);    // batch index

    const int dst = edge_dst[e];
    const int src = edge_src[e];
    const int gd = b * N_ + dst;
    const int gs = b * N_ + src;

    const float score = sd[gd] + ss[gs] + ab[0];
    const float w = __expf(-lrelu(score));

    if (lane == 0) atomicAdd(&rowsum[gd], w);

    const float* __restrict__ whs = Wh  + (size_t)gs * OUT_DIM;
    float* __restrict__       agd = agg + (size_t)gd * OUT_DIM;
    atomicAdd(&agd[lane],      w * whs[lane]);
    atomicAdd(&agd[lane + 32], w * whs[lane + 32]);
}

// ---------------------------------------------------------------------------
// Kernel 5: normalize, add residual, final leaky-relu.
// ---------------------------------------------------------------------------
__global__ __launch_bounds__(256) void gat_finalize_kernel(
    const float* __restrict__ agg, const float* __restrict__ rowsum,
    const float* __restrict__ res, float* __restrict__ out)
{
    const int i = blockIdx.x * blockDim.x + threadIdx.x;
    if (i >= BN * OUT_DIM) return;
    const int rowi = i >> 6;
    const float h = agg[i] / (rowsum[rowi] + EPS_) + res[i];
    out[i] = lrelu(h);
}

// ---------------------------------------------------------------------------
// Host-side launch. Inputs (setup_inputs order):
//  0:X 1:Ww 2:Wb 3:aw 4:ab 5:rw 6:rb 7:edge_dst 8:edge_src
// ---------------------------------------------------------------------------
extern "C" void kernel_launch(void* const* d_in, const int* in_sizes, int n_in,
                              void* d_out, int out_size, void* d_ws, size_t ws_size,
                              hipStream_t stream)
{
    const float* X        = (const float*)d_in[0];
    const float* Ww       = (const float*)d_in[1];
    const float* Wb       = (const float*)d_in[2];
    const float* aw       = (const float*)d_in[3];
    const float* ab       = (const float*)d_in[4];
    const float* rw       = (const float*)d_in[5];
    const float* rb       = (const float*)d_in[6];
    const int*   edge_dst = (const int*)d_in[7];
    const int*   edge_src = (const int*)d_in[8];
    float* out = (float*)d_out;

    // Workspace carve-up (floats): Wh | res | sd | ss | rowsum | agg
    float* Wh     = (float*)d_ws;                 // BN*64
    float* res    = Wh     + (size_t)BN * OUT_DIM;
    float* sd     = res    + (size_t)BN * OUT_DIM;
    float* ss     = sd     + BN;
    float* rowsum = ss     + BN;
    float* agg    = rowsum + BN;                  // BN*64

    // 1) Dense GEMMs on the WMMA pipe.
    gat_gemm_kernel<<<BN / 16, 256, 0, stream>>>(X, Ww, Wb, rw, rb, Wh, res);

    // 2) Per-node score dots (also zeroes rowsum).
    gat_node_score_kernel<<<(BN + 255) / 256, 256, 0, stream>>>(Wh, aw, sd, ss, rowsum);

    // 3) Zero agg.
    gat_zero_kernel<<<(BN * OUT_DIM + 255) / 256, 256, 0, stream>>>(agg, BN * OUT_DIM);

    // 4) Edge scatter: E*B waves, 8 waves per 256-thread block -> exact grid.
    gat_edge_kernel<<<(int)(((long long)E_ * B_) / 8), 256, 0, stream>>>(
        edge_dst, edge_src, sd, ss, ab, Wh, rowsum, agg);

    // 5) Finalize.
    gat_finalize_kernel<<<(BN * OUT_DIM + 255) / 256, 256, 0, stream>>>(agg, rowsum, res, out);
}